// ClusterMerging_14620068675723
// MI455X (gfx1250) — compile-verified
//
#include <hip/hip_runtime.h>
#include <hip/hip_bf16.h>
#include <math.h>

#define BB 4
#define NN 16384
#define CC 128
#define NBHD 48
#define TABLE 3969
#define INNER 4
#define OUTC 256
#define KEEP 4096
#define LNC 512
#define EPSLN 1e-5f

typedef __attribute__((ext_vector_type(2))) float v2f;
typedef __attribute__((ext_vector_type(8))) float v8f;

__device__ __forceinline__ unsigned ford(float v) {
    unsigned u = __float_as_uint(v);
    return (u & 0x80000000u) ? ~u : (u | 0x80000000u);
}

// Descending bitonic sort of NN 64-bit keys in LDS, 1024 threads.
__device__ __forceinline__ void bitonic_desc(unsigned long long* keys, int t) {
    for (unsigned k = 2; k <= (unsigned)NN; k <<= 1) {
        for (unsigned j = k >> 1; j > 0; j >>= 1) {
            __syncthreads();
            #pragma unroll
            for (int e = 0; e < NN / 2 / 1024; ++e) {
                unsigned tt = (unsigned)t + (unsigned)e * 1024u;
                unsigned i = 2u * tt - (tt & (j - 1u));
                unsigned l = i ^ j;
                unsigned long long a = keys[i], b = keys[l];
                bool up = ((i & k) == 0u);
                if (up ? (a < b) : (a > b)) { keys[i] = b; keys[l] = a; }
            }
        }
    }
    __syncthreads();
}

// One block per batch. Replicates jax.lax.top_k ordering exactly:
// descending value, ties broken by ascending index.
extern "C" __global__ __launch_bounds__(1024)
void topk_sort_kernel(const float* __restrict__ pos,
                      const float* __restrict__ learned_prob,
                      const int* __restrict__ stride_p,
                      const int* __restrict__ reserve_p,
                      int* __restrict__ idx_out) {
    extern __shared__ unsigned long long keys[];  // NN entries = 128 KB
    const int b = blockIdx.x;
    const int t = threadIdx.x;
    const int mod = stride_p[0] * 2;
    const int reserve_num = reserve_p[0];
    const int sample_num = KEEP - reserve_num;

    // Phase 1: final_prob = learned_prob + reserve_mask * (-100)
    for (int e = 0; e < NN / 1024; ++e) {
        int i = t + e * 1024;
        float v = learned_prob[b * NN + i];
        int px = (int)pos[(b * NN + i) * 2 + 0];
        int py = (int)pos[(b * NN + i) * 2 + 1];
        bool res = ((px % mod) == 0) && ((py % mod) == 0);
        float fv = v + (res ? -100.0f : 0.0f);
        keys[i] = (((unsigned long long)ford(fv)) << 32) |
                  (unsigned long long)(0xFFFFFFFFu - (unsigned)i);
    }
    bitonic_desc(keys, t);
    for (int r = t; r < sample_num; r += 1024)
        idx_out[b * KEEP + r] = (int)(0xFFFFFFFFu - (unsigned)(keys[r] & 0xFFFFFFFFull));
    __syncthreads();

    // Phase 2: top_k on the {0,1} reserve mask (ties -> ascending index)
    for (int e = 0; e < NN / 1024; ++e) {
        int i = t + e * 1024;
        int px = (int)pos[(b * NN + i) * 2 + 0];
        int py = (int)pos[(b * NN + i) * 2 + 1];
        bool res = ((px % mod) == 0) && ((py % mod) == 0);
        keys[i] = (((unsigned long long)(res ? 1u : 0u)) << 32) |
                  (unsigned long long)(0xFFFFFFFFu - (unsigned)i);
    }
    bitonic_desc(keys, t);
    for (int r = t; r < reserve_num; r += 1024)
        idx_out[b * KEEP + sample_num + r] =
            (int)(0xFFFFFFFFu - (unsigned)(keys[r] & 0xFFFFFFFFull));
}

// weights_table = GELU(LN(pre_table @ w1 + b1)), exact (erf) gelu. [3969,4]
extern "C" __global__ void wtab_kernel(const float* __restrict__ pre_table,
                                       const float* __restrict__ w1,
                                       const float* __restrict__ b1,
                                       const float* __restrict__ g1,
                                       const float* __restrict__ be1,
                                       float* __restrict__ wtab) {
    int r = blockIdx.x * blockDim.x + threadIdx.x;
    if (r >= TABLE) return;
    float p[5];
    #pragma unroll
    for (int j = 0; j < 5; ++j) p[j] = pre_table[r * 5 + j];
    float tv[4];
    #pragma unroll
    for (int i = 0; i < 4; ++i) {
        float s = b1[i];
        #pragma unroll
        for (int j = 0; j < 5; ++j) s += p[j] * w1[j * 4 + i];
        tv[i] = s;
    }
    float m = 0.25f * (tv[0] + tv[1] + tv[2] + tv[3]);
    float var = 0.0f;
    #pragma unroll
    for (int i = 0; i < 4; ++i) { float d = tv[i] - m; var += d * d; }
    var *= 0.25f;
    float rs = rsqrtf(var + EPSLN);
    #pragma unroll
    for (int i = 0; i < 4; ++i) {
        float y = (tv[i] - m) * rs * g1[i] + be1[i];
        wtab[r * 4 + i] = 0.5f * y * (1.0f + erff(y * 0.70710678118654752f));
    }
}

// pos_down = pos[idx]
extern "C" __global__ void pos_gather_kernel(const float* __restrict__ pos,
                                             const int* __restrict__ idx,
                                             float* __restrict__ out) {
    int t = blockIdx.x * blockDim.x + threadIdx.x;
    if (t >= BB * KEEP) return;
    int b = t >> 12;
    int i = idx[t];
    out[t * 2 + 0] = pos[(b * NN + i) * 2 + 0];
    out[t * 2 + 1] = pos[(b * NN + i) * 2 + 1];
}

// One 128-thread block per kept row: gather 48 neighbors, weighted reduce
// into [C,INNER], fused LayerNorm(512), write normalized row to X.
extern "C" __global__ __launch_bounds__(128)
void gather_reduce_ln_kernel(const float* __restrict__ feat,
                             const int* __restrict__ member_idx,
                             const float* __restrict__ cluster_mask,
                             const float* __restrict__ learned_prob,
                             const int* __restrict__ pe_idx,
                             const float* __restrict__ wtab,
                             const int* __restrict__ idx,
                             const float* __restrict__ gn,
                             const float* __restrict__ bn,
                             float* __restrict__ X) {
    __shared__ int s_m[NBHD];
    __shared__ float s_w[NBHD * INNER];
    __shared__ float s_red[CC];
    __shared__ int s_i;
    const int bid = blockIdx.x;       // b*KEEP + r
    const int b = bid >> 12;
    const int c = threadIdx.x;
    if (c == 0) s_i = idx[bid];
    __syncthreads();
    const int i = s_i;
    if (c < NBHD) {
        int base = (b * NN + i) * NBHD + c;
        int m = member_idx[base];
        int pe = pe_idx[base];
        float cm = cluster_mask[base];
        float s = learned_prob[b * NN + m] * cm;
        s_m[c] = m;
        s_w[c * 4 + 0] = wtab[pe * 4 + 0] * s;
        s_w[c * 4 + 1] = wtab[pe * 4 + 1] * s;
        s_w[c * 4 + 2] = wtab[pe * 4 + 2] * s;
        s_w[c * 4 + 3] = wtab[pe * 4 + 3] * s;
    }
    __syncthreads();
    float a0 = 0.f, a1 = 0.f, a2 = 0.f, a3 = 0.f;
    #pragma unroll 4
    for (int k = 0; k < NBHD; ++k) {
        int m = s_m[k];
        float f = feat[((b * NN + m) << 7) + c];
        a0 += s_w[k * 4 + 0] * f;
        a1 += s_w[k * 4 + 1] * f;
        a2 += s_w[k * 4 + 2] * f;
        a3 += s_w[k * 4 + 3] * f;
    }
    // LayerNorm over 512 (two-pass, matches reference mean/var)
    s_red[c] = a0 + a1 + a2 + a3;
    __syncthreads();
    for (int off = 64; off > 0; off >>= 1) {
        if (c < off) s_red[c] += s_red[c + off];
        __syncthreads();
    }
    float mean = s_red[0] * (1.0f / 512.0f);
    __syncthreads();
    float d0 = a0 - mean, d1 = a1 - mean, d2 = a2 - mean, d3 = a3 - mean;
    s_red[c] = d0 * d0 + d1 * d1 + d2 * d2 + d3 * d3;
    __syncthreads();
    for (int off = 64; off > 0; off >>= 1) {
        if (c < off) s_red[c] += s_red[c + off];
        __syncthreads();
    }
    float rs = rsqrtf(s_red[0] * (1.0f / 512.0f) + EPSLN);
    int base = bid * LNC + c * 4;
    X[base + 0] = d0 * rs * gn[c * 4 + 0] + bn[c * 4 + 0];
    X[base + 1] = d1 * rs * gn[c * 4 + 1] + bn[c * 4 + 1];
    X[base + 2] = d2 * rs * gn[c * 4 + 2] + bn[c * 4 + 2];
    X[base + 3] = d3 * rs * gn[c * 4 + 3] + bn[c * 4 + 3];
}

// out[16384,256] = X[16384,512] @ wl[512,256] + bl, f32 WMMA 16x16x4.
// One wave per 16x16 tile; 8 waves (8 row-tiles) per 256-thread block.
extern "C" __global__ __launch_bounds__(256)
void gemm_wmma_kernel(const float* __restrict__ X,
                      const float* __restrict__ wl,
                      const float* __restrict__ bl,
                      float* __restrict__ outF) {
    const int wave = threadIdx.x >> 5;
    const int lane = threadIdx.x & 31;
    const int row0 = (blockIdx.y * 8 + wave) * 16;
    const int col0 = blockIdx.x * 16;
    const int rr = lane & 15;
    const int half = lane >> 4;

    v8f acc = {};
    // A 16x4 layout: lanes 0-15 hold K=(0,1) in (v0,v1); lanes 16-31 hold K=(2,3)
    const float* Arow = X + (row0 + rr) * 512 + 2 * half;
    // B 4x16 layout mirrored: lane rr = column; rows (2*half, 2*half+1)
    const float* Bbase = wl + (2 * half) * OUTC + col0 + rr;

    #pragma unroll 4
    for (int k0 = 0; k0 < 512; k0 += 4) {
        v2f a = *(const v2f*)(Arow + k0);            // 8-byte load
        v2f bfrag;
        bfrag.x = Bbase[k0 * OUTC];
        bfrag.y = Bbase[k0 * OUTC + OUTC];
        acc = __builtin_amdgcn_wmma_f32_16x16x4_f32(
            false, a, false, bfrag, (short)0, acc, false, false);
    }
    float bias = bl[col0 + rr];
    #pragma unroll
    for (int v = 0; v < 8; ++v) {
        int m = row0 + (half ? v + 8 : v);
        outF[m * OUTC + col0 + rr] = acc[v] + bias;
    }
}

extern "C" void kernel_launch(void* const* d_in, const int* in_sizes, int n_in,
                              void* d_out, int out_size, void* d_ws, size_t ws_size,
                              hipStream_t stream) {
    (void)in_sizes; (void)n_in; (void)out_size; (void)ws_size;
    const float* pos          = (const float*)d_in[0];
    const float* feat         = (const float*)d_in[1];
    const int*   member_idx   = (const int*)d_in[2];
    const float* cluster_mask = (const float*)d_in[3];
    const float* learned_prob = (const float*)d_in[4];
    const int*   pe_idx       = (const int*)d_in[5];
    const float* pre_table    = (const float*)d_in[6];
    const float* w1           = (const float*)d_in[7];
    const float* b1           = (const float*)d_in[8];
    const float* g1           = (const float*)d_in[9];
    const float* be1          = (const float*)d_in[10];
    const float* gn           = (const float*)d_in[11];
    const float* bn           = (const float*)d_in[12];
    const float* wl           = (const float*)d_in[13];
    const float* bl           = (const float*)d_in[14];
    const int*   stride_p     = (const int*)d_in[15];
    const int*   reserve_p    = (const int*)d_in[16];

    char* ws = (char*)d_ws;
    int*   d_idx  = (int*)ws;                       // 64 KB
    float* d_wtab = (float*)(ws + 65536);           // 64 KB
    float* d_X    = (float*)(ws + 131072);          // 32 MB

    float* pos_out  = (float*)d_out;                // [4,4096,2]
    float* feat_out = pos_out + BB * KEEP * 2;      // [4,4096,256]

    topk_sort_kernel<<<BB, 1024, NN * sizeof(unsigned long long), stream>>>(
        pos, learned_prob, stride_p, reserve_p, d_idx);
    wtab_kernel<<<(TABLE + 255) / 256, 256, 0, stream>>>(
        pre_table, w1, b1, g1, be1, d_wtab);
    pos_gather_kernel<<<(BB * KEEP + 255) / 256, 256, 0, stream>>>(
        pos, d_idx, pos_out);
    gather_reduce_ln_kernel<<<BB * KEEP, 128, 0, stream>>>(
        feat, member_idx, cluster_mask, learned_prob, pe_idx,
        d_wtab, d_idx, gn, bn, d_X);
    gemm_wmma_kernel<<<dim3(OUTC / 16, (BB * KEEP) / (16 * 8)), 256, 0, stream>>>(
        d_X, wl, bl, feat_out);
}